// EncoderCoreDecoder_33500744909345
// MI455X (gfx1250) — compile-verified
//
#include <hip/hip_runtime.h>
#include <hip/hip_bf16.h>

// ---------------------------------------------------------------------------
// GraphNet encode-process-decode for MI455X (gfx1250), fp32 end-to-end.
// All GEMMs use V_WMMA_F32_16X16X4_F32 (one 16-row tile per wave32).
// ---------------------------------------------------------------------------

typedef __attribute__((ext_vector_type(2))) float v2f;
typedef __attribute__((ext_vector_type(8))) float v8f;

struct MLP { const float* w1; const float* b1; const float* w2; const float* b2; const float* g; const float* beta; };
struct Lin { const float* w; const float* b; };

#define WPB   4        // waves per block
#define BLOCK 128
#define LN_EPS 1e-5f

__device__ __forceinline__ v8f wmma4(v2f a, v2f b, v8f c) {
  // D = A(16x4,f32) * B(4x16,f32) + C(16x16,f32)
  return __builtin_amdgcn_wmma_f32_16x16x4_f32(false, a, false, b, (short)0, c, false, false);
}

// C += X_tile(16 x KN) * W(KN x 16).  arow = this lane's row base (row = lane&15).
template<int KN>
__device__ __forceinline__ void gemm_rows(const float* __restrict__ arow,
                                          const float* __restrict__ W,
                                          int lane, v8f& c) {
  const int half = lane >> 4;
  const int n    = lane & 15;
  #pragma unroll
  for (int k0 = 0; k0 < KN; k0 += 4) {
    v2f a, b;
    a.x = arow[k0 + 2 * half];
    a.y = arow[k0 + 2 * half + 1];
    b.x = W[(k0 + 2 * half) * 16 + n];
    b.y = W[(k0 + 2 * half + 1) * 16 + n];
    c = wmma4(a, b, c);
  }
}

// C0,C1 += H(16x16, LDS, stride ls) * W2(16x32)   (two 16-col output tiles)
__device__ __forceinline__ void gemm16_to32(const float* __restrict__ l, int ls,
                                            const float* __restrict__ W2,
                                            int lane, v8f& c0, v8f& c1) {
  const int half = lane >> 4;
  const int n    = lane & 15;
  const int m    = lane & 15;
  #pragma unroll
  for (int k0 = 0; k0 < 16; k0 += 4) {
    v2f a;  a.x  = l[m * ls + k0 + 2 * half];       a.y  = l[m * ls + k0 + 2 * half + 1];
    v2f b0; b0.x = W2[(k0 + 2 * half) * 32 + n];    b0.y = W2[(k0 + 2 * half + 1) * 32 + n];
    v2f b1; b1.x = W2[(k0 + 2 * half) * 32 + 16+n]; b1.y = W2[(k0 + 2 * half + 1) * 32 + 16+n];
    c0 = wmma4(a, b0, c0);
    c1 = wmma4(a, b1, c1);
  }
}

// C += X(16x32, LDS, stride ls) * W(32x16)
__device__ __forceinline__ void gemm32_to16(const float* __restrict__ l, int ls,
                                            const float* __restrict__ W,
                                            int lane, v8f& c) {
  const int half = lane >> 4;
  const int n    = lane & 15;
  const int m    = lane & 15;
  #pragma unroll
  for (int k0 = 0; k0 < 32; k0 += 4) {
    v2f a; a.x = l[m * ls + k0 + 2 * half]; a.y = l[m * ls + k0 + 2 * half + 1];
    v2f b; b.x = W[(k0 + 2 * half) * 16 + n]; b.y = W[(k0 + 2 * half + 1) * 16 + n];
    c = wmma4(a, b, c);
  }
}

// bias + ReLU on a 16x16 D-tile, stage into LDS (row-major, stride ls)
__device__ __forceinline__ void relu_bias_stage(const v8f& c, const float* __restrict__ b1,
                                                float* __restrict__ l, int ls, int lane) {
  const int half = lane >> 4;
  const int n    = lane & 15;
  const float bias = b1[n];
  #pragma unroll
  for (int r = 0; r < 8; r++) {
    float h = c[r] + bias;
    h = h > 0.f ? h : 0.f;
    l[(r + 8 * half) * ls + n] = h;
  }
}

// bias + ReLU + row-wise LayerNorm over 32 channels (two 16x16 D-tiles)
__device__ __forceinline__ void bias_relu_ln(const v8f& c0, const v8f& c1,
                                             const float* __restrict__ b2,
                                             const float* __restrict__ g,
                                             const float* __restrict__ beta,
                                             int lane, float o0[8], float o1[8]) {
  const int n = lane & 15;
  const float b2a = b2[n], b2b = b2[16 + n];
  const float ga = g[n],  gb = g[16 + n];
  const float ba = beta[n], bb = beta[16 + n];
  #pragma unroll
  for (int r = 0; r < 8; r++) {
    float h0 = c0[r] + b2a; h0 = h0 > 0.f ? h0 : 0.f;
    float h1 = c1[r] + b2b; h1 = h1 > 0.f ? h1 : 0.f;
    float s  = h0 + h1;
    float sq = h0 * h0 + h1 * h1;
    #pragma unroll
    for (int m = 1; m < 16; m <<= 1) {   // reduce within each 16-lane half
      s  += __shfl_xor(s,  m, 32);
      sq += __shfl_xor(sq, m, 32);
    }
    const float mu  = s * (1.f / 32.f);
    const float var = sq * (1.f / 32.f) - mu * mu;
    const float inv = rsqrtf(var + LN_EPS);
    o0[r] = (h0 - mu) * inv * ga + ba;
    o1[r] = (h1 - mu) * inv * gb + bb;
  }
}

// ---------------------------------------------------------------------------
// Kernels
// ---------------------------------------------------------------------------

// encoder: X(ntiles*16 x 16) -> MLP(16->16->32)+LN -> out and out0
__global__ __launch_bounds__(BLOCK) void gn_encode(const float* __restrict__ X, int ntiles,
                                                   MLP p, float* __restrict__ out,
                                                   float* __restrict__ out0) {
  __shared__ float smem[WPB][16 * 17];
  const int lane = threadIdx.x & 31;
  const int wid  = threadIdx.x >> 5;
  const int tile = blockIdx.x * WPB + wid;
  if (tile >= ntiles) return;
  const int row  = lane & 15;
  const int half = lane >> 4;
  const int n    = lane & 15;

  v8f c = {};
  gemm_rows<16>(X + ((size_t)tile * 16 + row) * 16, p.w1, lane, c);
  relu_bias_stage(c, p.b1, smem[wid], 17, lane);
  __builtin_amdgcn_wave_barrier();

  v8f c0 = {}, c1 = {};
  gemm16_to32(smem[wid], 17, p.w2, lane, c0, c1);
  float o0[8], o1[8];
  bias_relu_ln(c0, c1, p.b2, p.g, p.beta, lane, o0, o1);

  float* orow  = out  + (size_t)tile * 16 * 32;
  float* orow0 = out0 + (size_t)tile * 16 * 32;
  #pragma unroll
  for (int r = 0; r < 8; r++) {
    const int rr = r + 8 * half;
    orow[rr * 32 + n]       = o0[r];
    orow[rr * 32 + 16 + n]  = o1[r];
    orow0[rr * 32 + n]      = o0[r];
    orow0[rr * 32 + 16 + n] = o1[r];
  }
}

// core edge update: e_in = [e0,e, v0[s],v[s], v0[r],v[r]] (192) -> MLP(192->16->32)+LN -> e (in place)
__global__ __launch_bounds__(BLOCK) void gn_edge_update(const float* __restrict__ e0,
                                                        float* __restrict__ e,
                                                        const float* __restrict__ v0,
                                                        const float* __restrict__ v,
                                                        const int* __restrict__ senders,
                                                        const int* __restrict__ receivers,
                                                        int ntiles, MLP p) {
  __shared__ float smem[WPB][16 * 17];
  const int lane = threadIdx.x & 31;
  const int wid  = threadIdx.x >> 5;
  const int tile = blockIdx.x * WPB + wid;
  if (tile >= ntiles) return;
  const int row  = lane & 15;
  const int half = lane >> 4;
  const int n    = lane & 15;

  const size_t eidx = (size_t)tile * 16 + row;
  const int s = senders[eidx];
  const int r = receivers[eidx];

  v8f c = {};
  gemm_rows<32>(e0 + eidx * 32,     p.w1 + 0   * 16, lane, c);
  gemm_rows<32>(e  + eidx * 32,     p.w1 + 32  * 16, lane, c);
  gemm_rows<32>(v0 + (size_t)s*32,  p.w1 + 64  * 16, lane, c);
  gemm_rows<32>(v  + (size_t)s*32,  p.w1 + 96  * 16, lane, c);
  gemm_rows<32>(v0 + (size_t)r*32,  p.w1 + 128 * 16, lane, c);
  gemm_rows<32>(v  + (size_t)r*32,  p.w1 + 160 * 16, lane, c);
  relu_bias_stage(c, p.b1, smem[wid], 17, lane);
  __builtin_amdgcn_wave_barrier();

  v8f c0 = {}, c1 = {};
  gemm16_to32(smem[wid], 17, p.w2, lane, c0, c1);
  float o0[8], o1[8];
  bias_relu_ln(c0, c1, p.b2, p.g, p.beta, lane, o0, o1);

  float* orow = e + (size_t)tile * 16 * 32;
  #pragma unroll
  for (int rr8 = 0; rr8 < 8; rr8++) {
    const int rr = rr8 + 8 * half;
    orow[rr * 32 + n]      = o0[rr8];
    orow[rr * 32 + 16 + n] = o1[rr8];
  }
}

// scatter-add edges into per-receiver sums and counts
__global__ __launch_bounds__(256) void gn_agg(const float* __restrict__ e,
                                              const int* __restrict__ receivers,
                                              float* __restrict__ aggsum,
                                              float* __restrict__ cnt, int nedges) {
  const long long i = (long long)blockIdx.x * blockDim.x + threadIdx.x;
  if (i >= (long long)nedges * 32) return;
  const int edge = (int)(i >> 5);
  const int ch   = (int)(i & 31);
  const int r    = receivers[edge];
  atomicAdd(&aggsum[(size_t)r * 32 + ch], e[i]);
  if (ch == 0) atomicAdd(&cnt[r], 1.0f);
}

// mean = sum / max(cnt,1), in place
__global__ __launch_bounds__(256) void gn_agg_fin(float* __restrict__ aggsum,
                                                  const float* __restrict__ cnt, int nnodes) {
  const long long i = (long long)blockIdx.x * blockDim.x + threadIdx.x;
  if (i >= (long long)nnodes * 32) return;
  aggsum[i] = aggsum[i] / fmaxf(cnt[i >> 5], 1.0f);
}

// core node update: [v0,v,agg] (96) -> MLP(96->16->32)+LN -> v (in place)
__global__ __launch_bounds__(BLOCK) void gn_node_update(const float* __restrict__ v0,
                                                        float* __restrict__ v,
                                                        const float* __restrict__ agg,
                                                        int ntiles, MLP p) {
  __shared__ float smem[WPB][16 * 17];
  const int lane = threadIdx.x & 31;
  const int wid  = threadIdx.x >> 5;
  const int tile = blockIdx.x * WPB + wid;
  if (tile >= ntiles) return;
  const int row  = lane & 15;
  const int half = lane >> 4;
  const int n    = lane & 15;

  const size_t nidx = (size_t)tile * 16 + row;
  v8f c = {};
  gemm_rows<32>(v0  + nidx * 32, p.w1 + 0  * 16, lane, c);
  gemm_rows<32>(v   + nidx * 32, p.w1 + 32 * 16, lane, c);
  gemm_rows<32>(agg + nidx * 32, p.w1 + 64 * 16, lane, c);
  relu_bias_stage(c, p.b1, smem[wid], 17, lane);
  __builtin_amdgcn_wave_barrier();

  v8f c0 = {}, c1 = {};
  gemm16_to32(smem[wid], 17, p.w2, lane, c0, c1);
  float o0[8], o1[8];
  bias_relu_ln(c0, c1, p.b2, p.g, p.beta, lane, o0, o1);

  float* orow = v + (size_t)tile * 16 * 32;
  #pragma unroll
  for (int r = 0; r < 8; r++) {
    const int rr = r + 8 * half;
    orow[rr * 32 + n]      = o0[r];
    orow[rr * 32 + 16 + n] = o1[r];
  }
}

// decoder: X(32) -> MLP(32->16->32)+LN -> Linear(32->16) -> out
__global__ __launch_bounds__(BLOCK) void gn_decode(const float* __restrict__ X, int ntiles,
                                                   MLP p, Lin po, float* __restrict__ out) {
  __shared__ float smem[WPB][16 * 33];   // reused: stride 17 for H, stride 33 for LN result
  const int lane = threadIdx.x & 31;
  const int wid  = threadIdx.x >> 5;
  const int tile = blockIdx.x * WPB + wid;
  if (tile >= ntiles) return;
  const int row  = lane & 15;
  const int half = lane >> 4;
  const int n    = lane & 15;

  v8f c = {};
  gemm_rows<32>(X + ((size_t)tile * 16 + row) * 32, p.w1, lane, c);
  relu_bias_stage(c, p.b1, smem[wid], 17, lane);
  __builtin_amdgcn_wave_barrier();

  v8f c0 = {}, c1 = {};
  gemm16_to32(smem[wid], 17, p.w2, lane, c0, c1);
  float o0[8], o1[8];
  bias_relu_ln(c0, c1, p.b2, p.g, p.beta, lane, o0, o1);
  __builtin_amdgcn_wave_barrier();

  #pragma unroll
  for (int r = 0; r < 8; r++) {
    const int rr = r + 8 * half;
    smem[wid][rr * 33 + n]      = o0[r];
    smem[wid][rr * 33 + 16 + n] = o1[r];
  }
  __builtin_amdgcn_wave_barrier();

  v8f co = {};
  gemm32_to16(smem[wid], 33, po.w, lane, co);
  const float bo = po.b[n];
  float* orow = out + (size_t)tile * 16 * 16;
  #pragma unroll
  for (int r = 0; r < 8; r++) {
    const int rr = r + 8 * half;
    orow[rr * 16 + n] = co[r] + bo;
  }
}

// ---------------------------------------------------------------------------
// Host launch
// ---------------------------------------------------------------------------

extern "C" void kernel_launch(void* const* d_in, const int* in_sizes, int n_in,
                              void* d_out, int out_size, void* d_ws, size_t ws_size,
                              hipStream_t stream) {
  auto F = [&](int i) { return (const float*)d_in[i]; };

  const float *nodes, *edges;
  const int *senders, *receivers;
  MLP enc_node, enc_edge, core_edge, core_node, dec_node, dec_edge;
  Lin dec_node_out, dec_edge_out;
  int N, E;

  if (in_sizes[0] == 12800000) {
    // JAX pytree (sorted dict keys): edges, nodes, params(sorted), receivers, senders
    // mlp leaves sorted: beta, g, l1.b, l1.w, l2.b, l2.w ; linear: b, w
    edges = F(0); nodes = F(1);
    E = in_sizes[0] / 16; N = in_sizes[1] / 16;
    auto mlp = [&](int b) { return MLP{ F(b + 3), F(b + 2), F(b + 5), F(b + 4), F(b + 1), F(b + 0) }; };
    core_edge    = mlp(2);
    core_node    = mlp(8);
    dec_edge     = mlp(14);
    dec_edge_out = Lin{ F(21), F(20) };
    dec_node     = mlp(22);
    dec_node_out = Lin{ F(29), F(28) };
    enc_edge     = mlp(30);
    enc_node     = mlp(36);
    receivers = (const int*)d_in[42];
    senders   = (const int*)d_in[43];
  } else {
    // insertion order: nodes, edges, senders, receivers, params(insertion)
    // mlp leaves: l1.w, l1.b, l2.w, l2.b, g, beta ; linear: w, b
    nodes = F(0); edges = F(1);
    N = in_sizes[0] / 16; E = in_sizes[1] / 16;
    senders   = (const int*)d_in[2];
    receivers = (const int*)d_in[3];
    auto mlp = [&](int b) { return MLP{ F(b + 0), F(b + 1), F(b + 2), F(b + 3), F(b + 4), F(b + 5) }; };
    enc_node     = mlp(4);
    enc_edge     = mlp(10);
    core_edge    = mlp(16);
    core_node    = mlp(22);
    dec_node     = mlp(28);
    dec_node_out = Lin{ F(34), F(35) };
    dec_edge     = mlp(36);
    dec_edge_out = Lin{ F(42), F(43) };
  }

  // workspace layout (floats)
  float* ws = (float*)d_ws;
  float* v   = ws;                                  // N*32
  float* v0_ = v   + (size_t)N * 32;                // N*32
  float* e   = v0_ + (size_t)N * 32;                // E*32
  float* e0_ = e   + (size_t)E * 32;                // E*32
  float* agg = e0_ + (size_t)E * 32;                // N*32
  float* cnt = agg + (size_t)N * 32;                // N

  const int ntilesN = N / 16;   // 3125
  const int ntilesE = E / 16;   // 50000
  const dim3 blk(BLOCK);
  const dim3 gN((ntilesN + WPB - 1) / WPB);
  const dim3 gE((ntilesE + WPB - 1) / WPB);

  // encoder
  gn_encode<<<gN, blk, 0, stream>>>(nodes, ntilesN, enc_node, v, v0_);
  gn_encode<<<gE, blk, 0, stream>>>(edges, ntilesE, enc_edge, e, e0_);

  // core steps
  const long long aggElems = (long long)N * 32;
  const long long eElems   = (long long)E * 32;
  for (int step = 0; step < 3; step++) {
    gn_edge_update<<<gE, blk, 0, stream>>>(e0_, e, v0_, v, senders, receivers, ntilesE, core_edge);
    hipMemsetAsync(agg, 0, ((size_t)N * 32 + N) * sizeof(float), stream);  // agg + cnt
    gn_agg<<<(unsigned)((eElems + 255) / 256), 256, 0, stream>>>(e, receivers, agg, cnt, E);
    gn_agg_fin<<<(unsigned)((aggElems + 255) / 256), 256, 0, stream>>>(agg, cnt, N);
    gn_node_update<<<gN, blk, 0, stream>>>(v0_, v, agg, ntilesN, core_node);
  }

  // decoder: d_out = [v_out (N*16) | e_out (E*16)]
  float* out = (float*)d_out;
  gn_decode<<<gN, blk, 0, stream>>>(v, ntilesN, dec_node, dec_node_out, out);
  gn_decode<<<gE, blk, 0, stream>>>(e, ntilesE, dec_edge, dec_edge_out, out + (size_t)N * 16);
}